// EvolveGCN_H_41403484733449
// MI455X (gfx1250) — compile-verified
//
#include <hip/hip_runtime.h>
#include <math.h>

#define NN 100000
#define EE 1600000
#define FD 128

typedef float v2f __attribute__((ext_vector_type(2)));
typedef float v8f __attribute__((ext_vector_type(8)));

__device__ __forceinline__ void atomAddF(float* p, float v) {
  unsafeAtomicAdd(p, v);   // hardware global_atomic_add_f32 on gfx1250
}

// ---------------- p-norm ----------------
__global__ void pnorm_kernel(const float* __restrict__ p, float* __restrict__ pnorm) {
  __shared__ float red[128];
  int t = threadIdx.x;
  float v = p[t];
  red[t] = v * v;
  __syncthreads();
  for (int off = 64; off; off >>= 1) {
    if (t < off) red[t] += red[t + off];
    __syncthreads();
  }
  if (t == 0) pnorm[0] = sqrtf(red[0]);
}

// ---------------- degree ----------------
__global__ void zero_deg_kernel(unsigned* __restrict__ deg) {
  int i = blockIdx.x * blockDim.x + threadIdx.x;
  if (i < NN) deg[i] = 0u;
}

__global__ void deg_count_kernel(const int* __restrict__ ei, unsigned* __restrict__ deg) {
  int e = blockIdx.x * blockDim.x + threadIdx.x;
  if (e < EE) atomicAdd(&deg[ei[EE + e]], 1u);
}

__global__ void deg_to_dinv_kernel(float* __restrict__ dinv) {
  int i = blockIdx.x * blockDim.x + threadIdx.x;
  if (i < NN) {
    unsigned c = ((unsigned*)dinv)[i];
    dinv[i] = rsqrtf((float)c + 1.0f);
  }
}

// ---------------- scores: wave per node ----------------
__global__ void scores_kernel(const float* __restrict__ x, const float* __restrict__ p,
                              const float* __restrict__ pnorm, float* __restrict__ scores) {
  int gid  = blockIdx.x * blockDim.x + threadIdx.x;
  int node = gid >> 5;
  int lane = threadIdx.x & 31;
  if (node >= NN) return;
  const float* xr = x + (size_t)node * FD;
  float s = 0.0f;
#pragma unroll
  for (int m = 0; m < 4; ++m) s += xr[lane + 32 * m] * p[lane + 32 * m];
#pragma unroll
  for (int off = 16; off; off >>= 1) s += __shfl_xor(s, off, 32);
  if (lane == 0) scores[node] = s / pnorm[0];
}

// ---------------- top-k stage 1: bitonic sort of 512-chunks ----------------
__global__ void topk_stage1_kernel(const float* __restrict__ scores,
                                   float* __restrict__ cvals, int* __restrict__ cidx) {
  __shared__ float sv[512];
  __shared__ int   si[512];
  int c = blockIdx.x, t = threadIdx.x;
  for (int i = t; i < 512; i += 256) {
    int g = c * 512 + i;
    if (g < NN) { sv[i] = scores[g]; si[i] = g; }
    else        { sv[i] = -INFINITY; si[i] = 0x7fffffff; }
  }
  __syncthreads();
  for (int k = 2; k <= 512; k <<= 1) {
    for (int j = k >> 1; j > 0; j >>= 1) {
      for (int i = t; i < 512; i += 256) {
        int ixj = i ^ j;
        if (ixj > i) {
          float va = sv[i], vb = sv[ixj];
          int   ia = si[i], ib = si[ixj];
          bool aFirst = (va > vb) || (va == vb && ia < ib);  // "a before b" (descending)
          bool dirDesc = ((i & k) == 0);
          bool doSwap = dirDesc ? !aFirst : aFirst;
          if (doSwap) { sv[i] = vb; sv[ixj] = va; si[i] = ib; si[ixj] = ia; }
        }
      }
      __syncthreads();
    }
  }
  if (t < 128) { cvals[c * 128 + t] = sv[t]; cidx[c * 128 + t] = si[t]; }
}

// ---------------- top-k stage 2: 128 sequential arg-max selections ----------------
__global__ void __launch_bounds__(1024, 1)
topk_stage2_kernel(const float* __restrict__ cvals, const int* __restrict__ cidx, int M,
                   int* __restrict__ top_idx, float* __restrict__ top_gate) {
  int t = threadIdx.x;
  float v[25];
  int   id[25];
#pragma unroll
  for (int j = 0; j < 25; ++j) {
    int p = t + j * 1024;
    if (p < M) { v[j] = cvals[p]; id[j] = cidx[p]; }
    else       { v[j] = -INFINITY; id[j] = 0x7fffffff; }
  }
  __shared__ float sv[1024];
  __shared__ int   si[1024];
  __shared__ int   sp[1024];
  for (int it = 0; it < 128; ++it) {
    float bv = -INFINITY; int bi = 0x7fffffff; int bp = 0;
#pragma unroll
    for (int j = 0; j < 25; ++j) {
      bool better = (v[j] > bv) || (v[j] == bv && id[j] < bi);
      if (better) { bv = v[j]; bi = id[j]; bp = j; }
    }
    sv[t] = bv; si[t] = bi; sp[t] = t + bp * 1024;
    __syncthreads();
    for (int off = 512; off; off >>= 1) {
      if (t < off) {
        float ov = sv[t + off]; int oi = si[t + off];
        if ((ov > sv[t]) || (ov == sv[t] && oi < si[t])) {
          sv[t] = ov; si[t] = oi; sp[t] = sp[t + off];
        }
      }
      __syncthreads();
    }
    int   pos = sp[0];
    float wv  = sv[0];
    int   wi  = si[0];
    if (t == 0) { top_idx[it] = wi; top_gate[it] = tanhf(wv); }
#pragma unroll
    for (int j = 0; j < 25; ++j)
      if (pos == t + j * 1024) v[j] = -INFINITY;
    __syncthreads();
  }
}

// ---------------- fused GRU step -> W_new ----------------
__global__ void gru_kernel(const float* __restrict__ x, const float* __restrict__ W,
                           const float* __restrict__ w_ih, const float* __restrict__ w_hh,
                           const float* __restrict__ b_ih, const float* __restrict__ b_hh,
                           const int* __restrict__ top_idx, const float* __restrict__ top_gate,
                           float* __restrict__ Wn) {
  __shared__ float sx[FD];
  __shared__ float sw[FD];
  int i = blockIdx.x, j = threadIdx.x;
  int src = top_idx[i];
  float g = top_gate[i];
  sx[j] = x[(size_t)src * FD + j] * g;
  sw[j] = W[i * FD + j];
  __syncthreads();
  float gir = b_ih[j], giz = b_ih[j + FD], gin = b_ih[j + 2 * FD];
  float ghr = b_hh[j], ghz = b_hh[j + FD], ghn = b_hh[j + 2 * FD];
  const float* wr = w_ih + (size_t)j * FD;
  const float* wz = w_ih + (size_t)(j + FD) * FD;
  const float* wn = w_ih + (size_t)(j + 2 * FD) * FD;
  const float* hr = w_hh + (size_t)j * FD;
  const float* hz = w_hh + (size_t)(j + FD) * FD;
  const float* hn = w_hh + (size_t)(j + 2 * FD) * FD;
#pragma unroll 4
  for (int k = 0; k < FD; ++k) {
    float a = sx[k], b = sw[k];
    gir += a * wr[k]; giz += a * wz[k]; gin += a * wn[k];
    ghr += b * hr[k]; ghz += b * hz[k]; ghn += b * hn[k];
  }
  float r = 1.0f / (1.0f + expf(-(gir + ghr)));
  float z = 1.0f / (1.0f + expf(-(giz + ghz)));
  float n = tanhf(gin + r * ghn);
  Wn[i * FD + j] = (1.0f - z) * n + z * sw[j];
}

// ---------------- xw = x @ W_new via V_WMMA_F32_16X16X4_F32 ----------------
// 8 waves per block; wave w owns 16-column tile w; B fragments preloaded once,
// grid-stride over 16-row tiles.
__global__ void gemm_xw_wmma_kernel(const float* __restrict__ x, const float* __restrict__ Wn,
                                    float* __restrict__ xw, int nRowTiles) {
  int wave    = threadIdx.x >> 5;
  int lane    = threadIdx.x & 31;
  int colBase = wave * 16;
  int halfSel = lane >> 4;   // 0: K pair {0,1}; 1: K pair {2,3}
  int l15     = lane & 15;

  v2f bfrag[32];
#pragma unroll
  for (int ks = 0; ks < 32; ++ks) {
    int k0 = ks * 4 + halfSel * 2;
    bfrag[ks].x = Wn[(size_t)(k0 + 0) * FD + colBase + l15];
    bfrag[ks].y = Wn[(size_t)(k0 + 1) * FD + colBase + l15];
  }

  for (int rt = blockIdx.x; rt < nRowTiles; rt += gridDim.x) {
    const float* xrow = x + ((size_t)rt * 16 + l15) * FD;
    v8f c = {};
#pragma unroll
    for (int ks = 0; ks < 32; ++ks) {
      int k0 = ks * 4 + halfSel * 2;
      v2f a;
      a.x = xrow[k0 + 0];
      a.y = xrow[k0 + 1];
      c = __builtin_amdgcn_wmma_f32_16x16x4_f32(false, a, false, bfrag[ks],
                                                (short)0, c, false, false);
    }
    float* outp = xw + (size_t)rt * 16 * FD + colBase;
#pragma unroll
    for (int vv = 0; vv < 8; ++vv) {
      int m = halfSel * 8 + vv;
      outp[(size_t)m * FD + l15] = c[vv];
    }
  }
}

// ---------------- agg init: self-loop + bias ----------------
__global__ void agg_init_kernel(const float* __restrict__ xw, const float* __restrict__ dinv,
                                const float* __restrict__ bias, float* __restrict__ agg) {
  long long gid = (long long)blockIdx.x * blockDim.x + threadIdx.x;
  if (gid >= (long long)NN * FD) return;
  int i = (int)(gid >> 7);
  int f = (int)(gid & 127);
  float d = dinv[i];
  agg[gid] = xw[gid] * d * d + bias[f];
}

// ---------------- edge scatter: wave per edge ----------------
__global__ void scatter_kernel(const int* __restrict__ ei, const float* __restrict__ xw,
                               const float* __restrict__ dinv, float* __restrict__ agg) {
  long long gid = (long long)blockIdx.x * blockDim.x + threadIdx.x;
  int e    = (int)(gid >> 5);
  int lane = (int)(gid & 31);
  if (e >= EE) return;
  int r = ei[e];
  int c = ei[EE + e];
  float coef = dinv[r] * dinv[c];
  const float4* src = (const float4*)(xw + (size_t)r * FD);
  float* dst = agg + (size_t)c * FD;
  float4 v = src[lane];
  atomAddF(&dst[lane * 4 + 0], v.x * coef);
  atomAddF(&dst[lane * 4 + 1], v.y * coef);
  atomAddF(&dst[lane * 4 + 2], v.z * coef);
  atomAddF(&dst[lane * 4 + 3], v.w * coef);
}

// ---------------- head: relu + linear, wave per node ----------------
__global__ void head_kernel(const float* __restrict__ agg, const float* __restrict__ lin_w,
                            const float* __restrict__ lin_b, float* __restrict__ out) {
  int gid  = blockIdx.x * blockDim.x + threadIdx.x;
  int node = gid >> 5;
  int lane = threadIdx.x & 31;
  if (node >= NN) return;
  const float* ar = agg + (size_t)node * FD;
  float s = 0.0f;
#pragma unroll
  for (int m = 0; m < 4; ++m) {
    float a = ar[lane + 32 * m];
    s += (a > 0.0f ? a : 0.0f) * lin_w[lane + 32 * m];
  }
#pragma unroll
  for (int off = 16; off; off >>= 1) s += __shfl_xor(s, off, 32);
  if (lane == 0) out[node] = s + lin_b[0];
}

extern "C" void kernel_launch(void* const* d_in, const int* in_sizes, int n_in,
                              void* d_out, int out_size, void* d_ws, size_t ws_size,
                              hipStream_t stream) {
  const float* x        = (const float*)d_in[0];
  const int*   ei       = (const int*)d_in[1];
  const float* W        = (const float*)d_in[2];
  const float* p        = (const float*)d_in[3];
  const float* w_ih     = (const float*)d_in[4];
  const float* w_hh     = (const float*)d_in[5];
  const float* b_ih     = (const float*)d_in[6];
  const float* b_hh     = (const float*)d_in[7];
  const float* cbias    = (const float*)d_in[8];
  const float* lin_w    = (const float*)d_in[9];
  const float* lin_b    = (const float*)d_in[10];
  float* out = (float*)d_out;

  // workspace layout (256B aligned slices)
  char* base = (char*)d_ws;
  size_t off = 0;
  auto take = [&](size_t bytes) -> char* {
    char* ptr = base + off;
    off = (off + bytes + 255) & ~(size_t)255;
    return ptr;
  };
  float* xw       = (float*)take((size_t)NN * FD * 4);
  float* agg      = (float*)take((size_t)NN * FD * 4);
  float* scores   = (float*)take((size_t)NN * 4);
  float* dinv     = (float*)take((size_t)NN * 4);
  const int NCH   = (NN + 511) / 512;          // 196 chunks
  const int M     = NCH * 128;                 // 25088 candidates
  float* cvals    = (float*)take((size_t)M * 4);
  int*   cidx     = (int*)take((size_t)M * 4);
  int*   top_idx  = (int*)take(128 * 4);
  float* top_gate = (float*)take(128 * 4);
  float* pnorm    = (float*)take(256);
  float* Wn       = (float*)take((size_t)FD * FD * 4);
  (void)ws_size; (void)in_sizes; (void)n_in; (void)out_size;

  // 1. ||p||
  pnorm_kernel<<<1, 128, 0, stream>>>(p, pnorm);
  // 2. degrees -> dinv (store counts in dinv buffer as u32, convert in place)
  zero_deg_kernel<<<(NN + 255) / 256, 256, 0, stream>>>((unsigned*)dinv);
  deg_count_kernel<<<(EE + 255) / 256, 256, 0, stream>>>(ei, (unsigned*)dinv);
  deg_to_dinv_kernel<<<(NN + 255) / 256, 256, 0, stream>>>(dinv);
  // 3. scores (wave per node)
  scores_kernel<<<(NN * 32 + 255) / 256, 256, 0, stream>>>(x, p, pnorm, scores);
  // 4. exact top-128
  topk_stage1_kernel<<<NCH, 256, 0, stream>>>(scores, cvals, cidx);
  topk_stage2_kernel<<<1, 1024, 0, stream>>>(cvals, cidx, M, top_idx, top_gate);
  // 5. GRU -> W_new
  gru_kernel<<<FD, FD, 0, stream>>>(x, W, w_ih, w_hh, b_ih, b_hh, top_idx, top_gate, Wn);
  // 6. xw = x @ W_new (fp32 WMMA)
  gemm_xw_wmma_kernel<<<512, 256, 0, stream>>>(x, Wn, xw, NN / 16);
  // 7. agg = xw * dinv^2 + bias
  agg_init_kernel<<<(NN * FD + 255) / 256, 256, 0, stream>>>(xw, dinv, cbias, agg);
  // 8. edge scatter (wave per edge, f32 hardware atomics)
  scatter_kernel<<<((size_t)EE * 32 + 255) / 256, 256, 0, stream>>>(ei, xw, dinv, agg);
  // 9. head
  head_kernel<<<(NN * 32 + 255) / 256, 256, 0, stream>>>(agg, lin_w, lin_b, out);
}